// LSTM_4_75333726372407
// MI455X (gfx1250) — compile-verified
//
#include <hip/hip_runtime.h>

// Problem constants (from reference): H=21, T=8192, 4H=84 gate rows.
static constexpr int kT = 8192;
static constexpr int kH = 21;
static constexpr int kG = 84;   // 4*H
static constexpr int kK3 = 50;  // LSTM3 input width (o1:1 + o2:49)

typedef __attribute__((ext_vector_type(2))) float v2f;
typedef __attribute__((ext_vector_type(8))) float v8f;

// Single-trans-op activations (v_exp_f32 is base-2 on AMD; v_rcp_f32).
__device__ __forceinline__ float fast_sigmoid(float x) {
  float e = __builtin_amdgcn_exp2f(-1.4426950408889634f * x);
  return __builtin_amdgcn_rcpf(1.0f + e);
}
__device__ __forceinline__ float fast_tanh(float x) {
  float e = __builtin_amdgcn_exp2f(2.8853900817779268f * x);
  return 1.0f - 2.0f * __builtin_amdgcn_rcpf(e + 1.0f);
}

// ---------------------------------------------------------------------------
// Phase A: G1[t,j] = b1[j] + in1[t]·Wih1[j,:], G2 likewise. K=3 -> pure VALU,
// memory-shaped; fully parallel over (t, j).
// ---------------------------------------------------------------------------
__global__ void gates12_kernel(const float* __restrict__ src,
                               const float* __restrict__ Wih1, const float* __restrict__ b1,
                               const float* __restrict__ Wih2, const float* __restrict__ b2,
                               float* __restrict__ G1, float* __restrict__ G2) {
  int idx = blockIdx.x * blockDim.x + threadIdx.x;
  if (idx >= kT * kG) return;
  int t = idx / kG, j = idx % kG;
  float s0 = src[t * 4 + 0], s1 = src[t * 4 + 1];
  float s2 = src[t * 4 + 2], s3 = src[t * 4 + 3];
  // in1 = (x1, x3a, x3b) ; in2 = (x2, x3a, x3b)
  G1[idx] = b1[j] + Wih1[j * 3 + 0] * s0 + Wih1[j * 3 + 1] * s2 + Wih1[j * 3 + 2] * s3;
  G2[idx] = b2[j] + Wih2[j * 3 + 0] * s1 + Wih2[j * 3 + 1] * s2 + Wih2[j * 3 + 2] * s3;
}

// ---------------------------------------------------------------------------
// Sequential LSTM scan core: one wave32 per LSTM. Lane L owns gate rows
// {L, L+32, L+64(<84)} of Whh (weights resident in VGPRs). h is broadcast via
// LDS. PROJ = number of output-projection rows fused after each step.
// ---------------------------------------------------------------------------
template <int PROJ>
__device__ __forceinline__ void lstm_scan_core(
    const float* __restrict__ G,     // [kT, kG] precomputed x-gates (+bias)
    const float* __restrict__ Whh,   // [kG, kH]
    const float* __restrict__ Wp,    // [PROJ, kH]
    const float* __restrict__ bp,    // [PROJ]
    float* __restrict__ outbuf, int out_stride, int out_off) {
  __shared__ float hbuf[32];
  __shared__ float zbuf[kG];
  const int L = threadIdx.x;

  float wA[kH], wB[kH], wC[kH];
#pragma unroll
  for (int k = 0; k < kH; ++k) {
    wA[k] = Whh[L * kH + k];
    wB[k] = Whh[(L + 32) * kH + k];
    wC[k] = (L < kG - 64) ? Whh[(L + 64) * kH + k] : 0.0f;
  }
  const bool p0 = (L < PROJ), p1 = (L + 32 < PROJ);
  float wP0[kH], wP1[kH];
#pragma unroll
  for (int k = 0; k < kH; ++k) {
    wP0[k] = p0 ? Wp[L * kH + k] : 0.0f;
    wP1[k] = p1 ? Wp[(L + 32) * kH + k] : 0.0f;
  }
  const float bp0 = p0 ? bp[L] : 0.0f;
  const float bp1 = p1 ? bp[L + 32] : 0.0f;

  float h[kH];
#pragma unroll
  for (int k = 0; k < kH; ++k) h[k] = 0.0f;
  float c = 0.0f;
  hbuf[L] = 0.0f;
  __syncthreads();

  for (int t = 0; t < kT; ++t) {
    const float* g = G + t * kG;
    __builtin_prefetch(g + kG, 0, 0);  // global_prefetch_b8 for next step
    float a0 = g[L];
    float a1 = g[L + 32];
    float a2 = (L < kG - 64) ? g[L + 64] : 0.0f;
#pragma unroll
    for (int k = 0; k < kH; ++k) {
      a0 = fmaf(wA[k], h[k], a0);
      a1 = fmaf(wB[k], h[k], a1);
      a2 = fmaf(wC[k], h[k], a2);
    }
    zbuf[L] = a0;
    zbuf[L + 32] = a1;
    if (L < kG - 64) zbuf[L + 64] = a2;
    __syncthreads();  // single-wave WG: cheap s_wait_dscnt
    if (L < kH) {
      float ig = fast_sigmoid(zbuf[L]);
      float fg = fast_sigmoid(zbuf[kH + L]);
      float gg = fast_tanh(zbuf[2 * kH + L]);
      float og = fast_sigmoid(zbuf[3 * kH + L]);
      c = fg * c + ig * gg;
      hbuf[L] = og * fast_tanh(c);
    }
    __syncthreads();
#pragma unroll
    for (int k = 0; k < kH; ++k) h[k] = hbuf[k];
    if (PROJ > 0) {  // fused output projection: lane L -> rows L, L+32
      float o0 = bp0, o1 = bp1;
#pragma unroll
      for (int k = 0; k < kH; ++k) {
        o0 = fmaf(wP0[k], h[k], o0);
        o1 = fmaf(wP1[k], h[k], o1);
      }
      if (p0) outbuf[t * out_stride + out_off + L] = o0;
      if (p1) outbuf[t * out_stride + out_off + L + 32] = o1;
    }
  }
}

// Phase B: LSTM1 and LSTM2 run concurrently (grid=2), both write in3[t,50].
__global__ void scan12_kernel(const float* __restrict__ G1, const float* __restrict__ Whh1,
                              const float* __restrict__ W1, const float* __restrict__ bl1,
                              const float* __restrict__ G2, const float* __restrict__ Whh2,
                              const float* __restrict__ W2, const float* __restrict__ bl2,
                              float* __restrict__ in3) {
  if (blockIdx.x == 0)
    lstm_scan_core<1>(G1, Whh1, W1, bl1, in3, kK3, 0);    // o1 -> in3[:,0]
  else
    lstm_scan_core<49>(G2, Whh2, W2, bl2, in3, kK3, 1);   // o2 -> in3[:,1:50]
}

// Phase D: LSTM3 scan, fused final projection to d_out [kT, 2].
__global__ void scan3_kernel(const float* __restrict__ G3, const float* __restrict__ Whh3,
                             const float* __restrict__ W3, const float* __restrict__ bl3,
                             float* __restrict__ out) {
  lstm_scan_core<2>(G3, Whh3, W3, bl3, out, 2, 0);
}

// ---------------------------------------------------------------------------
// Phase C: G3[8192x84] = in3[8192x50] @ Wih3^T[50x84] + b3 via
// V_WMMA_F32_16X16X4_F32 (fp32 in/out -> no precision loss).
// A tile 16x4 layout (ISA 7.12.2): lane<16 -> {K0,K1}, lane>=16 -> {K2,K3};
// B tile 4x16 mirrored: VGPR0 rows {0,2}, VGPR1 rows {1,3} across lane halves.
// C/D: VGPR v, lane L -> row t0+v+8*(L/16), col j0+(L%16).
//
// All loads are UNCONDITIONAL from clamped (always-valid) addresses.
// Column padding (jrow >= 84) needs NO masking: D[m,n] = sum_k A[m,k]*B[k,n],
// so clamped-row B garbage only lands in columns the store guard discards.
// Only the K tail (k=50,51 -> chunk 12, hi lane half) must be zeroed, since
// invalid k terms would pollute every output; that costs one select+mul in
// the final chunk only. EXEC stays full; loads merge into b64s off one base.
// ---------------------------------------------------------------------------
__global__ void gates3_wmma_kernel(const float* __restrict__ X,     // [kT, 50]
                                   const float* __restrict__ Wih3,  // [84, 50]
                                   const float* __restrict__ b3,    // [84]
                                   float* __restrict__ G3) {        // [kT, 84]
  const int t0 = blockIdx.x * 16;   // 512 M-tiles
  const int j0 = blockIdx.y * 16;   // 6 N-tiles (84 -> pad 96)
  const int lane = threadIdx.x;
  const int half = lane >> 4;
  const int m = lane & 15;
  const int trow = t0 + m;
  const int jrow = j0 + m;
  const int jrow_c = (jrow < kG) ? jrow : (kG - 1);    // clamped, always valid

  const float* __restrict__ xrow = X + (size_t)trow * kK3;
  const float* __restrict__ wrow = Wih3 + (size_t)jrow_c * kK3;

  v8f acc = {};
#pragma unroll
  for (int kc = 0; kc < 13; ++kc) {   // K = 50, padded to 52
    const int k0 = kc * 4 + 2 * half;
    const int k0c = (k0 <= kK3 - 2) ? k0 : (kK3 - 2);  // folds away for kc<12
    v2f a; a[0] = xrow[k0c]; a[1] = xrow[k0c + 1];     // global_load_b64
    float b0 = wrow[k0c], b1 = wrow[k0c + 1];          // global_load_b64
    if (kc == 12) {  // zero the invalid k=50,51 terms (hi lane half only)
      const float kmask = (k0 <= kK3 - 2) ? 1.0f : 0.0f;
      b0 *= kmask;
      b1 *= kmask;
    }
    v2f b; b[0] = b0; b[1] = b1;
    acc = __builtin_amdgcn_wmma_f32_16x16x4_f32(
        /*neg_a=*/false, a, /*neg_b=*/false, b,
        /*c_mod=*/(short)0, acc, /*reuse_a=*/false, /*reuse_b=*/false);
  }
  if (jrow < kG) {
    float bias = b3[jrow];
#pragma unroll
    for (int v = 0; v < 8; ++v) {
      int row = t0 + v + 8 * half;
      G3[row * kG + jrow] = acc[v] + bias;
    }
  }
}

// ---------------------------------------------------------------------------
extern "C" void kernel_launch(void* const* d_in, const int* in_sizes, int n_in,
                              void* d_out, int out_size, void* d_ws, size_t ws_size,
                              hipStream_t stream) {
  (void)in_sizes; (void)n_in; (void)out_size; (void)ws_size;
  const float* src  = (const float*)d_in[0];
  const float* Wih1 = (const float*)d_in[1];
  const float* Whh1 = (const float*)d_in[2];
  const float* b1   = (const float*)d_in[3];
  const float* W1   = (const float*)d_in[4];
  const float* bl1  = (const float*)d_in[5];
  const float* Wih2 = (const float*)d_in[6];
  const float* Whh2 = (const float*)d_in[7];
  const float* b2   = (const float*)d_in[8];
  const float* W2   = (const float*)d_in[9];
  const float* bl2  = (const float*)d_in[10];
  const float* Wih3 = (const float*)d_in[11];
  const float* Whh3 = (const float*)d_in[12];
  const float* b3   = (const float*)d_in[13];
  const float* W3   = (const float*)d_in[14];
  const float* bl3  = (const float*)d_in[15];

  // Workspace layout (floats): G1, G2, G3 each kT*kG; in3 kT*50. ~9.9 MB.
  float* ws  = (float*)d_ws;
  float* G1  = ws;
  float* G2  = G1 + (size_t)kT * kG;
  float* G3  = G2 + (size_t)kT * kG;
  float* in3 = G3 + (size_t)kT * kG;

  // Stream ordering provides inter-phase global barriers (graph-capturable).
  gates12_kernel<<<(kT * kG + 255) / 256, 256, 0, stream>>>(src, Wih1, b1, Wih2, b2, G1, G2);
  scan12_kernel<<<2, 32, 0, stream>>>(G1, Whh1, W1, bl1, G2, Whh2, W2, bl2, in3);
  gates3_wmma_kernel<<<dim3(512, 6), 32, 0, stream>>>(in3, Wih3, b3, G3);
  scan3_kernel<<<1, 32, 0, stream>>>(G3, Whh3, W3, bl3, (float*)d_out);
}